// HAR_LSTM_6chan_10685878633195
// MI455X (gfx1250) — compile-verified
//
#include <hip/hip_runtime.h>

typedef float v2f __attribute__((ext_vector_type(2)));
typedef float v8f __attribute__((ext_vector_type(8)));

#define TT 2048
#define BB 1024
#define GG 12
#define HH 3

// fast sigmoid: 1 / (1 + e^-x)   (v_exp_f32 + v_rcp_f32)
__device__ __forceinline__ float fsig(float x) {
    return __builtin_amdgcn_rcpf(1.0f + __expf(-x));
}

// LSTM cell pointwise stage operating on WMMA D layout:
//   low lanes  (seq s): d[v] = gate v   -> i0,i1,i2,f0,f1,f2,g0,g1
//   high lanes (seq s): d[v] = gate 8+v -> g2,o0,o1,o2,(0..)
// tm/ta: per-lane-half constants so d[0] gets sigmoid (lo) or tanh (hi).
__device__ __forceinline__ void lstm_cell(const v8f& d, float tm, float ta,
                                          float& c0, float& c1, float& c2,
                                          float& h0, float& h1, float& h2) {
    float a0 = fmaf(fsig(d[0] * tm), tm, ta);          // lo: sig(i0)   hi: tanh(g2)
    float a1 = fsig(d[1]);                              // lo: sig(i1)   hi: sig(o0)
    float a2 = fsig(d[2]);                              // lo: sig(i2)   hi: sig(o1)
    float a3 = fsig(d[3]);                              // lo: sig(f0)   hi: sig(o2)
    float a4 = fsig(d[4]);                              // lo: sig(f1)
    float a5 = fsig(d[5]);                              // lo: sig(f2)
    float a6 = fmaf(fsig(d[6] * 2.f), 2.f, -1.f);       // lo: tanh(g0)
    float a7 = fmaf(fsig(d[7] * 2.f), 2.f, -1.f);       // lo: tanh(g1)
    // pull tanh(g2) and sig(o0..2) from the opposite lane half
    float sg2 = __shfl_xor(a0, 16, 32);
    float so0 = __shfl_xor(a1, 16, 32);
    float so1 = __shfl_xor(a2, 16, 32);
    float so2 = __shfl_xor(a3, 16, 32);
    c0 = fmaf(a3, c0, a0 * a6);
    c1 = fmaf(a4, c1, a1 * a7);
    c2 = fmaf(a5, c2, a2 * sg2);
    h0 = so0 * fmaf(fsig(c0 * 2.f), 2.f, -1.f);
    h1 = so1 * fmaf(fsig(c1 * 2.f), 2.f, -1.f);
    h2 = so2 * fmaf(fsig(c2 * 2.f), 2.f, -1.f);
}

// Pack h (valid in low lanes) into WMMA B layout: element (N=lane&15, K=vgpr+2*hi)
__device__ __forceinline__ v2f make_bh(float h0, float h1, float h2, bool hi) {
    float sh2 = __shfl_xor(h2, 16, 32);  // high lanes fetch low-lane h2 (K=2)
    v2f b;
    b[0] = hi ? sh2 : h0;   // K=0 / K=2
    b[1] = hi ? 0.f : h1;   // K=1 / K=3(pad)
    return b;
}

__global__ void __launch_bounds__(256)
har_prep_kernel(float* __restrict__ out, int n4, int out_size,
                const float* __restrict__ gamma, const float* __restrict__ beta,
                const float* __restrict__ mean, const float* __restrict__ var,
                float2* __restrict__ ab) {
    int i = blockIdx.x * blockDim.x + threadIdx.x;
    if (i < n4) {
        float4 z = make_float4(0.f, 0.f, 0.f, 0.f);
        reinterpret_cast<float4*>(out)[i] = z;
    }
    if (i == 0) {
        for (int r = n4 * 4; r < out_size; ++r) out[r] = 0.f;
    }
    if (i < TT) {
        float s = gamma[i] * rsqrtf(var[i] + 1e-5f);
        // fold BN + (1/3 encoder mean):  contrib = max(h*A + B, 0)
        ab[i] = make_float2(s * (1.f / 3.f), (beta[i] - mean[i] * s) * (1.f / 3.f));
    }
}

__global__ void __launch_bounds__(32)
har_lstm_wmma_kernel(const float* __restrict__ x,
                     const float* __restrict__ Wih0, const float* __restrict__ Whh0,
                     const float* __restrict__ bih0, const float* __restrict__ bhh0,
                     const float* __restrict__ Wih1, const float* __restrict__ Whh1,
                     const float* __restrict__ bih1, const float* __restrict__ bhh1,
                     const float2* __restrict__ ab, float* __restrict__ out) {
    const int lane = threadIdx.x;
    const bool hi  = lane >= 16;
    const int  s   = lane & 15;

    const int wid   = blockIdx.x;      // 0..383
    const int enc   = wid / 128;       // encoder 0..2
    const int tile  = wid % 128;       // 16-sequence tile
    const int chain = tile * 16 + s;   // 0..2047 (acc: 0..1023, gyr: 1024..2047)
    const int b     = (chain < BB) ? chain : chain - BB;
    const int cbase = (chain < BB) ? 0 : 3;

    // ---- A matrices (weights), layout (M = s, K = vgpr + 2*hi) ----
    const int  k0    = hi ? 2 : 0;
    const int  k1    = hi ? 3 : 1;
    const bool mok   = (s < GG);
    const bool k1ok  = mok && !hi;     // K=3 is pad
    const int  woff  = enc * GG * HH + s * HH;
    v2f aih0, ahh0, aih1, ahh1;
    aih0[0] = mok  ? Wih0[woff + k0] : 0.f;
    aih0[1] = k1ok ? Wih0[woff + k1] : 0.f;
    ahh0[0] = mok  ? Whh0[woff + k0] : 0.f;
    ahh0[1] = k1ok ? Whh0[woff + k1] : 0.f;
    aih1[0] = mok  ? Wih1[woff + k0] : 0.f;
    aih1[1] = k1ok ? Wih1[woff + k1] : 0.f;
    ahh1[0] = mok  ? Whh1[woff + k0] : 0.f;
    ahh1[1] = k1ok ? Whh1[woff + k1] : 0.f;

    // ---- bias carried in the C operand, layout (M = v + 8*hi, N = s) ----
    v8f cb0, cb1;
#pragma unroll
    for (int v = 0; v < 8; ++v) {
        int m = v + (hi ? 8 : 0);
        bool val = (m < GG);
        cb0[v] = val ? (bih0[enc * GG + m] + bhh0[enc * GG + m]) : 0.f;
        cb1[v] = val ? (bih1[enc * GG + m] + bhh1[enc * GG + m]) : 0.f;
    }

    // ---- per-lane input stream pointers (x is [B,6,T], t contiguous) ----
    const float* pA = x + ((size_t)b * 6 + cbase + (hi ? 2 : 0)) * TT; // K0/K2 channel
    const float* pB = x + ((size_t)b * 6 + cbase + 1) * TT;           // K1 channel (lo lanes)

    float* o0 = out + ((size_t)b * 6 + cbase + 0) * TT;
    float* o1 = out + ((size_t)b * 6 + cbase + 1) * TT;
    float* o2 = out + ((size_t)b * 6 + cbase + 2) * TT;

    // ---- state ----
    float c00 = 0.f, c01 = 0.f, c02 = 0.f, h00 = 0.f, h01 = 0.f, h02 = 0.f;
    float c10 = 0.f, c11 = 0.f, c12 = 0.f, h10 = 0.f, h11 = 0.f, h12 = 0.f;
    v2f bh0prev; bh0prev[0] = 0.f; bh0prev[1] = 0.f;
    v2f bh1prev; bh1prev[0] = 0.f; bh1prev[1] = 0.f;
    const float tm = hi ? 2.f : 1.f;
    const float ta = hi ? -1.f : 0.f;

    for (int t = 0; t < TT; t += 4) {
        const float4 fA = *reinterpret_cast<const float4*>(pA + t);
        const float4 fB = *reinterpret_cast<const float4*>(pB + t);
        if (t < TT - 256) {                     // stay in-bounds within the row
            __builtin_prefetch(pA + t + 256, 0, 1);   // global_prefetch_b8
            __builtin_prefetch(pB + t + 256, 0, 1);
        }
        const float xa[4] = {fA.x, fA.y, fA.z, fA.w};
        const float xb[4] = {fB.x, fB.y, fB.z, fB.w};

#pragma unroll
        for (int u = 0; u < 4; ++u) {
            // B operand for the input projection
            v2f bx;
            bx[0] = xa[u];
            bx[1] = hi ? 0.f : xb[u];

            // ---- layer 0: gates = W_hh0*h_prev + bias, += W_ih0*x_t ----
            v8f d0 = __builtin_amdgcn_wmma_f32_16x16x4_f32(
                false, ahh0, false, bh0prev, (short)0, cb0, false, false);
            d0 = __builtin_amdgcn_wmma_f32_16x16x4_f32(
                false, aih0, false, bx, (short)0, d0, false, false);
            lstm_cell(d0, tm, ta, c00, c01, c02, h00, h01, h02);
            v2f bh0 = make_bh(h00, h01, h02, hi);

            // ---- layer 1: input = layer-0 h of this step ----
            v8f d1 = __builtin_amdgcn_wmma_f32_16x16x4_f32(
                false, ahh1, false, bh1prev, (short)0, cb1, false, false);
            d1 = __builtin_amdgcn_wmma_f32_16x16x4_f32(
                false, aih1, false, bh0, (short)0, d1, false, false);
            lstm_cell(d1, tm, ta, c10, c11, c12, h10, h11, h12);
            v2f bh1 = make_bh(h10, h11, h12, hi);

            bh0prev = bh0;
            bh1prev = bh1;

            // ---- fused BN + ReLU + 1/3 encoder-mean, accumulate ----
            float2 sc = ab[t + u];
            float r0 = fmaxf(fmaf(h10, sc.x, sc.y), 0.f);
            float r1 = fmaxf(fmaf(h11, sc.x, sc.y), 0.f);
            float r2 = fmaxf(fmaf(h12, sc.x, sc.y), 0.f);
            if (!hi) {  // low lanes own real per-sequence h; reconverges before next WMMA
                __hip_atomic_fetch_add(o0 + t + u, r0, __ATOMIC_RELAXED, __HIP_MEMORY_SCOPE_AGENT);
                __hip_atomic_fetch_add(o1 + t + u, r1, __ATOMIC_RELAXED, __HIP_MEMORY_SCOPE_AGENT);
                __hip_atomic_fetch_add(o2 + t + u, r2, __ATOMIC_RELAXED, __HIP_MEMORY_SCOPE_AGENT);
            }
        }
    }
}

extern "C" void kernel_launch(void* const* d_in, const int* in_sizes, int n_in,
                              void* d_out, int out_size, void* d_ws, size_t ws_size,
                              hipStream_t stream) {
    const float* x    = (const float*)d_in[0];
    const float* Wih0 = (const float*)d_in[1];
    const float* Whh0 = (const float*)d_in[2];
    const float* bih0 = (const float*)d_in[3];
    const float* bhh0 = (const float*)d_in[4];
    const float* Wih1 = (const float*)d_in[5];
    const float* Whh1 = (const float*)d_in[6];
    const float* bih1 = (const float*)d_in[7];
    const float* bhh1 = (const float*)d_in[8];
    const float* gamma = (const float*)d_in[9];
    const float* beta  = (const float*)d_in[10];
    const float* mean  = (const float*)d_in[11];
    const float* var   = (const float*)d_in[12];
    float*  out = (float*)d_out;
    float2* ab  = (float2*)d_ws;   // 2048 * 8B = 16 KB of workspace

    int n4 = out_size / 4;
    int zblocks = (n4 + 255) / 256;
    har_prep_kernel<<<zblocks, 256, 0, stream>>>(out, n4, out_size, gamma, beta, mean, var, ab);

    // 128 sequence-tiles * 3 encoders = 384 waves, one wave per workgroup
    har_lstm_wmma_kernel<<<384, 32, 0, stream>>>(x, Wih0, Whh0, bih0, bhh0,
                                                 Wih1, Whh1, bih1, bhh1,
                                                 (const float2*)d_ws, out);
}